// Sampling_24352464569651
// MI455X (gfx1250) — compile-verified
//
#include <hip/hip_runtime.h>
#include <math.h>
#include <stdint.h>

// ---- problem constants (from the reference) --------------------------------
#define LAT      49
#define BATCH    100
#define COVROWS  (1000 * LAT)               // 49000
#define NCPLX    (COVROWS * LAT)            // 2,401,000 complex64 outputs
#define NV4      (NCPLX / 4)                // 600,250 float4 chunks of cov_noise
#define ZERO_V4  (((BATCH + LAT - 1) * LAT) / 4)  // 7252/4 = 1813 zeroed float4s
#define TILE     256                        // float4s per tile (one block-wide load)
#define NTILES   (NV4 / TILE)               // 2344 full tiles
#define NTAIL    (NV4 - NTILES * TILE)      // 186 tail float4s
#define NBLK     586                        // 2344 / 586 = 4 tiles per block
#define NY       (BATCH * LAT)              // 4900

typedef float v4f __attribute__((ext_vector_type(4)));

// ---------------------------------------------------------------------------
// Fused kernel.
//  * Bulk: float32 stream -> complex64 stream (imag = 0), rows 0..147 zeroed.
//    Double-buffered async global->LDS staging (ASYNCcnt), NT 128-bit stores.
//    Each lane round-trips only its own 16B LDS slot => no barriers needed.
//  * Y: first NY global threads compute
//    Y[b,l] = Re( round(DFT[l,l]) * sqrt(eig[99,l]) * (a[l,b] + i b[l,b]) ).
//    round(cos/sin(2*pi*k/49)), k = l^2 mod 49, is in {-1,0,1} with exact
//    integer thresholds (boundaries never hit .5), so no trig is needed.
// ---------------------------------------------------------------------------
__global__ __launch_bounds__(256) void sampling_fused_kernel(
    const v4f* __restrict__ in,       // cov_noise as float4 stream
    v4f* __restrict__ out,            // cov output as float4 stream
    const float* __restrict__ eig,    // [BATCH, LAT]
    const float* __restrict__ a_eps,  // [LAT, BATCH]
    const float* __restrict__ b_eps,  // [LAT, BATCH]
    float* __restrict__ yout) {       // [BATCH, LAT]
  __shared__ v4f buf[2][TILE];
  const int tid = threadIdx.x;

  // Wave-relative LDS byte offsets of this lane's two staging slots.
  // (Generic-pointer low 32 bits == addrspace(3) offset on gfx1250.)
  const uint32_t l0 = (uint32_t)(uintptr_t)&buf[0][tid];
  const uint32_t l1 = (uint32_t)(uintptr_t)&buf[1][tid];

  long t = blockIdx.x;
  const long stride = gridDim.x;
  int p = 0;

  // Kick off the first async tile load immediately.
  if (t < NTILES) {
    uint64_t g = (uint64_t)(uintptr_t)(in + t * TILE + tid);
    asm volatile("global_load_async_to_lds_b128 %0, %1, off"
                 :: "v"(l0), "v"(g) : "memory");
  }

  // --- Y pass (tiny, overlaps with in-flight async load) -------------------
  {
    const int gid = (int)blockIdx.x * 256 + tid;
    if (gid < NY) {
      const int b = gid / LAT;
      const int l = gid - b * LAT;
      const int k = (l * l) % LAT;
      // round(cos(2*pi*k/49)):  1 for k<=8 or k>=41; -1 for 17<=k<=32; else 0
      const float dr = (k <= 8 || k >= 41) ? 1.f
                     : (k >= 17 && k <= 32) ? -1.f : 0.f;
      // round(sin(2*pi*k/49)):  1 for 5<=k<=20; -1 for 29<=k<=44; else 0
      const float di = (k >= 5 && k <= 20) ? 1.f
                     : (k >= 29 && k <= 44) ? -1.f : 0.f;
      const float s  = sqrtf(eig[(BATCH - 1) * LAT + l]);
      const float ar = a_eps[l * BATCH + b];
      const float br = b_eps[l * BATCH + b];
      yout[gid] = (dr * s) * ar - (di * s) * br;
    }
  }

  // --- bulk expand loop -----------------------------------------------------
  for (; t < NTILES; t += stride) {
    const long tn = t + stride;
    if (tn < NTILES) {
      // Prefetch next tile into the other buffer, then wait for the older one.
      uint64_t g = (uint64_t)(uintptr_t)(in + tn * TILE + tid);
      asm volatile("global_load_async_to_lds_b128 %0, %1, off"
                   :: "v"(p ? l0 : l1), "v"(g) : "memory");
      asm volatile("s_wait_asynccnt 1" ::: "memory");
    } else {
      asm volatile("s_wait_asynccnt 0" ::: "memory");
    }

    v4f v = buf[p][tid];
    const long base = t * TILE + tid;
    if (base < (long)ZERO_V4) { v = (v4f){0.f, 0.f, 0.f, 0.f}; }

    const v4f lo = {v.x, 0.f, v.y, 0.f};
    const v4f hi = {v.z, 0.f, v.w, 0.f};
    __builtin_nontemporal_store(lo, out + 2 * base);
    __builtin_nontemporal_store(hi, out + 2 * base + 1);
    p ^= 1;
  }

  // Tail: 186 float4s, handled directly (all far past the zeroed region).
  if (blockIdx.x == 0 && tid < NTAIL) {
    const long base = (long)NTILES * TILE + tid;
    const v4f v = in[base];
    const v4f lo = {v.x, 0.f, v.y, 0.f};
    const v4f hi = {v.z, 0.f, v.w, 0.f};
    __builtin_nontemporal_store(lo, out + 2 * base);
    __builtin_nontemporal_store(hi, out + 2 * base + 1);
  }
}

// ---------------------------------------------------------------------------
extern "C" void kernel_launch(void* const* d_in, const int* in_sizes, int n_in,
                              void* d_out, int out_size, void* d_ws, size_t ws_size,
                              hipStream_t stream) {
  (void)in_sizes; (void)n_in; (void)out_size; (void)d_ws; (void)ws_size;

  const float* eigenvalues = (const float*)d_in[0];   // [100,49]
  const float* a_eps       = (const float*)d_in[1];   // [49,100]
  const float* b_eps       = (const float*)d_in[2];   // [49,100]
  const float* cov_noise   = (const float*)d_in[3];   // [49000,49]

  float* out = (float*)d_out;   // cov (complex64 interleaved), then Y

  sampling_fused_kernel<<<NBLK, 256, 0, stream>>>(
      (const v4f*)cov_noise, (v4f*)out,
      eigenvalues, a_eps, b_eps, out + 2 * (size_t)NCPLX);
}